// MultistreamLoss_10496900071736
// MI455X (gfx1250) — compile-verified
//
#include <hip/hip_runtime.h>

#define MARGIN 2.0f
#define BLK 256

typedef __attribute__((ext_vector_type(2))) float v2f;
typedef __attribute__((ext_vector_type(8))) float v8f;

// ---------------------------------------------------------------------------
// Kernel 1: tiled O(n^2) pairwise ranking loss.
// Grid (jb, ib): block handles j in [jb*256, jb*256+256), i-chunk ib.
// Only tiles with ib <= jb contribute (i < j); diagonal tile masks via loop
// bound = tid (branch-free). i-chunk (label, pred, 4 streams) staged in LDS,
// read back as broadcast ds_load (all lanes same address -> no conflicts).
// Deterministic block tree-reduction -> 3 partials per block in d_ws.
// ---------------------------------------------------------------------------
__global__ void __launch_bounds__(BLK)
pair_loss_kernel(const float* __restrict__ scores,
                 const float* __restrict__ pred,
                 const float* __restrict__ label,
                 float* __restrict__ partials,
                 int n, int S, int nblk)
{
    __shared__ float ls_label[BLK];
    __shared__ float ls_vec[5][BLK];   // 0 = pred, 1..4 = streams

    const int tid   = threadIdx.x;
    const int jb    = blockIdx.x;
    const int ib    = blockIdx.y;
    const int j     = jb * BLK + tid;
    const int ig    = ib * BLK + tid;

    // cooperative i-chunk load (zero padded past n; padded entries are never
    // selected because i >= n implies i >= j for any valid j)
    float zl = 0.0f, zp = 0.0f;
    if (ig < n) { zl = label[ig]; zp = pred[ig]; }
    ls_label[tid]  = zl;
    ls_vec[0][tid] = zp;
    #pragma unroll
    for (int k = 0; k < 4; ++k) {
        float v = 0.0f;
        if (k < S && ig < n) v = scores[(size_t)k * (size_t)n + ig];
        ls_vec[1 + k][tid] = v;
    }
    __syncthreads();

    // j-side values live in registers for the whole tile
    float lj = 0.0f;
    float vj[5] = {0.0f, 0.0f, 0.0f, 0.0f, 0.0f};
    if (j < n) {
        lj    = label[j];
        vj[0] = pred[j];
        #pragma unroll
        for (int k = 0; k < 4; ++k)
            if (k < S) vj[1 + k] = scores[(size_t)k * (size_t)n + j];
    }

    // iteration bound: full chunk below diagonal, tid on the diagonal
    int limit = 0;
    if (j < n) {
        if (ib < jb)       limit = BLK;
        else if (ib == jb) limit = tid;   // local i < tid  <=>  global i < j
    }

    float accS = 0.0f;   // sum over 4 streams of ranking terms
    float accP = 0.0f;   // pred ranking terms
    float accM = 0.0f;   // mse terms

    #pragma unroll 4
    for (int i = 0; i < limit; ++i) {
        float dl = lj - ls_label[i];
        float s  = (dl > 0.0f) ? 1.0f : ((dl < 0.0f) ? -1.0f : 0.0f);
        float ns = -s;
        // pred: relu(margin - (p_j - p_i) * s)
        {
            float t = fmaf(ns, vj[0] - ls_vec[0][i], MARGIN);
            accP += fmaxf(t, 0.0f);
        }
        #pragma unroll
        for (int k = 0; k < 4; ++k) {
            if (k < S) {
                float t = fmaf(ns, vj[1 + k] - ls_vec[1 + k][i], MARGIN);
                accS += fmaxf(t, 0.0f);
            }
        }
    }

    // MSE term counted exactly once per j (diagonal tiles only)
    if (ib == jb && j < n) {
        float d = vj[0] - lj;
        accM = d * d;
    }

    // deterministic block tree reduction
    __shared__ float r0[BLK], r1[BLK], r2[BLK];
    r0[tid] = accS; r1[tid] = accP; r2[tid] = accM;
    __syncthreads();
    for (int off = BLK >> 1; off > 0; off >>= 1) {
        if (tid < off) {
            r0[tid] += r0[tid + off];
            r1[tid] += r1[tid + off];
            r2[tid] += r2[tid + off];
        }
        __syncthreads();
    }
    if (tid == 0) {
        const int slot = jb * nblk + ib;      // tiles with ib > jb write zeros
        partials[3 * slot + 0] = r0[0];
        partials[3 * slot + 1] = r1[0];
        partials[3 * slot + 2] = r2[0];
    }
}

// ---------------------------------------------------------------------------
// Wave32 cross-lane sum via V_WMMA_F32_16X16X4_F32 (deterministic, RNE).
// Step 1: A = {x, 0} (A[m][0]=x[m], A[m][2]=x[m+16]), B = ones
//         -> D1[m][n] = x[m] + x[m+16]                      (1 WMMA)
// Step 2: A = ones, B taken from D1's VGPR pairs, C accumulates
//         -> C[m][n] = sum over all 16 row-values = total    (4 WMMAs)
// EXEC is all-ones: single uniform 32-thread block, no divergence here.
// ---------------------------------------------------------------------------
__device__ __forceinline__ float wave32_sum_wmma(float x)
{
    v2f A;    A[0] = x;    A[1] = 0.0f;
    v2f Ones; Ones[0] = 1.0f; Ones[1] = 1.0f;
    v8f Z = {};
    v8f D1 = __builtin_amdgcn_wmma_f32_16x16x4_f32(
        /*neg_a=*/false, A, /*neg_b=*/false, Ones,
        /*c_mod=*/(short)0, Z, /*reuse_a=*/false, /*reuse_b=*/false);

    v8f C = {};
    #pragma unroll
    for (int t = 0; t < 4; ++t) {
        v2f B; B[0] = D1[2 * t]; B[1] = D1[2 * t + 1];
        C = __builtin_amdgcn_wmma_f32_16x16x4_f32(
            false, Ones, false, B, (short)0, C, false, false);
    }
    return C[0];   // total replicated in every element/lane
}

__global__ void __launch_bounds__(32)
final_reduce_kernel(const float* __restrict__ partials,
                    float* __restrict__ out,
                    int nslots, int n, int S)
{
    const int lane = threadIdx.x;
    float a = 0.0f, b = 0.0f, c = 0.0f;
    for (int s = lane; s < nslots; s += 32) {
        a += partials[3 * s + 0];
        b += partials[3 * s + 1];
        c += partials[3 * s + 2];
    }
    const float sumS = wave32_sum_wmma(a);   // stream ranking total
    const float sumP = wave32_sum_wmma(b);   // pred ranking total
    const float sumM = wave32_sum_wmma(c);   // sum of squared errors

    if (lane == 0) {
        const float alpha = 1.0f / (float)S;                       // alpha_ms
        const float beta  = (n == 1) ? 1.0f
                                     : 2.0f / ((float)n * (float)(n - 1));
        out[0] = alpha * sumS + (sumM / (float)n) + beta * sumP;
    }
}

// ---------------------------------------------------------------------------
extern "C" void kernel_launch(void* const* d_in, const int* in_sizes, int n_in,
                              void* d_out, int out_size, void* d_ws, size_t ws_size,
                              hipStream_t stream)
{
    const float* scores = (const float*)d_in[0];   // [S, n]
    const float* pred   = (const float*)d_in[1];   // [n]
    const float* label  = (const float*)d_in[2];   // [n]

    const int n = in_sizes[1];
    const int S = (n > 0) ? (in_sizes[0] / n) : 0;
    const int nblk = (n + BLK - 1) / BLK;

    float* partials = (float*)d_ws;                // 3 * nblk^2 floats

    dim3 grid(nblk, nblk);
    pair_loss_kernel<<<grid, BLK, 0, stream>>>(scores, pred, label,
                                               partials, n, S, nblk);
    final_reduce_kernel<<<1, 32, 0, stream>>>(partials, (float*)d_out,
                                              nblk * nblk, n, S);
}